// H3AN_55877524521375
// MI455X (gfx1250) — compile-verified
//
#include <hip/hip_runtime.h>
#include <math.h>

typedef __attribute__((ext_vector_type(2))) float v2f;
typedef __attribute__((ext_vector_type(8))) float v8f;

// ---------------------------------------------------------------------------
// Kernel 1: row-orthonormalize the 16 Phi_P 3x3 blocks exactly as
// jnp.linalg.qr(Phi_P^T) -> Q^T would (Householder, beta = -sign(alpha)*||x||,
// the LAPACK/XLA convention). One thread per matrix; writes 16*9 floats.
// ---------------------------------------------------------------------------
__global__ void haar_qr16(const float* __restrict__ PhiP, float* __restrict__ phi) {
  int i = threadIdx.x;
  if (i >= 16) return;
  const float* P = PhiP + i * 9;          // P[r*3+c] = Phi_P[i][r][c]
  float T[3][3];                           // T = Phi_P[i]^T
#pragma unroll
  for (int r = 0; r < 3; ++r)
#pragma unroll
    for (int c = 0; c < 3; ++c) T[r][c] = P[c * 3 + r];

  // Householder reflector 1 (column 0)
  float tau1 = 0.f, v11 = 0.f, v12 = 0.f;
  {
    float x0 = T[0][0], x1 = T[1][0], x2 = T[2][0];
    float xn2 = x1 * x1 + x2 * x2;
    if (xn2 > 0.f) {
      float normx = sqrtf(x0 * x0 + xn2);
      float beta = (x0 >= 0.f) ? -normx : normx;
      tau1 = (beta - x0) / beta;
      float s = 1.f / (x0 - beta);
      v11 = x1 * s; v12 = x2 * s;
#pragma unroll
      for (int c = 0; c < 3; ++c) {       // T = H1 * T (needed for reflector 2)
        float w = T[0][c] + v11 * T[1][c] + v12 * T[2][c];
        T[0][c] -= tau1 * w;
        T[1][c] -= tau1 * w * v11;
        T[2][c] -= tau1 * w * v12;
      }
    }
  }
  // Householder reflector 2 (column 1, rows 1..2)
  float tau2 = 0.f, v22 = 0.f;
  {
    float y0 = T[1][1], y1 = T[2][1];
    float yn2 = y1 * y1;
    if (yn2 > 0.f) {
      float normy = sqrtf(y0 * y0 + yn2);
      float beta = (y0 >= 0.f) ? -normy : normy;
      tau2 = (beta - y0) / beta;
      float s = 1.f / (y0 - beta);
      v22 = y1 * s;
    }
  }
  // Q = H1 * H2 (applied to identity)
  float Q[3][3] = {{1.f, 0.f, 0.f}, {0.f, 1.f, 0.f}, {0.f, 0.f, 1.f}};
  Q[1][1] -= tau2;
  Q[1][2] -= tau2 * v22;
  Q[2][1] -= tau2 * v22;
  Q[2][2] -= tau2 * v22 * v22;
#pragma unroll
  for (int c = 0; c < 3; ++c) {
    float w = Q[0][c] + v11 * Q[1][c] + v12 * Q[2][c];
    Q[0][c] -= tau1 * w;
    Q[1][c] -= tau1 * w * v11;
    Q[2][c] -= tau1 * w * v12;
  }
  // Phi[i] = Q^T  (orthonormal rows)
#pragma unroll
  for (int j = 0; j < 3; ++j)
#pragma unroll
    for (int k = 0; k < 3; ++k) phi[i * 9 + j * 3 + k] = Q[k][j];
}

// ---------------------------------------------------------------------------
// Kernel 2: one butterfly level via V_WMMA_F32_16X16X4_F32.
//   A (16x4) = Phi rows 0..2 (zero-padded)     -> wave-invariant
//   B (4x16) = 16 consecutive groups of 3 f's  -> column n = group g0+n, K pad 0
//   D rows 0/1/2 = avg/d1/d2 for those 16 groups, contiguous in lanes 0..15.
// f32 layout per ISA: lanes 0-15 hold K0 (v0) / K1 (v1); lanes 16-31 K2/K3.
//
// Cache policy (192 MB L2, 23.3 TB/s HBM):
//   NT_IN=true  : level-0 input f is read exactly once -> NT loads, don't let
//                 the 172 MB stream evict the resident avg cascade (~86 MB).
//   details     : written once, never re-read -> NT stores.
//   avg chain   : written then re-read by the next level -> RT (L2-resident).
// ---------------------------------------------------------------------------
template <bool NT_IN>
__global__ void haar_level_wmma(const float* __restrict__ in,
                                float* __restrict__ out,      // d_out base
                                float* __restrict__ avg,      // next level input
                                const float* __restrict__ phi9,
                                long long G) {                // #groups = M
  const int lane = (int)(threadIdx.x & 31u);
  const long long wid =
      (long long)blockIdx.x * (blockDim.x >> 5) + (threadIdx.x >> 5);
  const long long g0 = wid * 16;
  if (g0 >= G) return;                     // wave-uniform exit

  const int m = lane & 15;                 // A row / B column / D column
  const bool hi = lane >= 16;              // K half selector

  // A = Phi (rows >= 3 and K=3 padded with 0)
  v2f a; a.x = 0.f; a.y = 0.f;
  if (m < 3) {
    a.x = phi9[m * 3 + (hi ? 2 : 0)];
    if (!hi) a.y = phi9[m * 3 + 1];
  }

  // B: column m = f[3*(g0+m) + k], k = K index for this lane half
  const long long g = g0 + m;
  v2f b; b.x = 0.f; b.y = 0.f;
  if (g < G) {
    const float* p = in + 3 * g;
    if (NT_IN) {
      if (hi) { b.x = __builtin_nontemporal_load(p + 2); }
      else    { b.x = __builtin_nontemporal_load(p + 0);
                b.y = __builtin_nontemporal_load(p + 1); }
    } else {
      if (hi) { b.x = p[2]; }
      else    { b.x = p[0]; b.y = p[1]; }
    }
  }

  v8f c = {};
  c = __builtin_amdgcn_wmma_f32_16x16x4_f32(false, a, false, b, (short)0, c,
                                            false, false);

  // D[0][n]=avg, D[1][n]=d1, D[2][n]=d2 live in c[0..2] on lanes 0..15 (n=lane)
  if (!hi && g < G) {
    const long long M = G;
    avg[g] = c[0];
    __builtin_nontemporal_store(c[1], out + M + g);
    __builtin_nontemporal_store(c[2], out + 2 * M + g);
  }
}

// ---------------------------------------------------------------------------
// Host side: 1 QR launch + 16 level launches, avg chain ping-pongs in d_ws.
// ws need: (3^15 + 3^14 + 144) floats ~= 73 MB.
// ---------------------------------------------------------------------------
extern "C" void kernel_launch(void* const* d_in, const int* in_sizes, int n_in,
                              void* d_out, int out_size, void* d_ws,
                              size_t ws_size, hipStream_t stream) {
  (void)in_sizes; (void)n_in; (void)out_size; (void)ws_size;
  const float* f    = (const float*)d_in[0];   // 3^16 floats
  const float* PhiP = (const float*)d_in[1];   // 16*3*3 floats
  float* out = (float*)d_out;
  float* ws  = (float*)d_ws;

  const long long P15 = 14348907LL;            // 3^15
  const long long P14 = 4782969LL;             // 3^14
  float* W0  = ws;                              // avg buffers (ping-pong)
  float* W1  = ws + P15;
  float* phi = ws + P15 + P14;                  // 144 floats of Phi

  haar_qr16<<<1, 16, 0, stream>>>(PhiP, phi);

  const float* in = f;
  long long G = P15;                            // groups at level 0
  for (int i = 0; i < 16; ++i) {
    float* avg = (i == 15) ? out : ((i & 1) ? W1 : W0);  // last avg -> d_out[0]
    long long tiles  = (G + 15) / 16;           // one 16-group tile per wave
    long long blocks = (tiles + 7) / 8;         // 8 waves per 256-thread block
    if (i == 0) {
      haar_level_wmma<true><<<dim3((unsigned)blocks), dim3(256), 0, stream>>>(
          in, out, avg, phi + i * 9, G);
    } else {
      haar_level_wmma<false><<<dim3((unsigned)blocks), dim3(256), 0, stream>>>(
          in, out, avg, phi + i * 9, G);
    }
    in = avg;
    G /= 3;
  }
}